// CustomMetaPath2Vec_47966194762445
// MI455X (gfx1250) — compile-verified
//
#include <hip/hip_runtime.h>

typedef __attribute__((ext_vector_type(2))) float v2f;
typedef __attribute__((ext_vector_type(8))) float v8f;

#define DIM 128
#define CTX 7
#define ROWS_PER_WAVE 16
#define WAVES_PER_BLOCK 4

__global__ void mp2v_zero_kernel(float* out) { out[0] = 0.0f; }

// One wave32 handles 16 walk rows. A = 16 start vectors (16xK), B = 16 context
// vectors (Kx16); D = A*B accumulated over K=128 via 32x V_WMMA_F32_16X16X4_F32.
// Diagonal D[i][i] = dot(start_i, ctx_i). Two contexts are processed per pass
// with independent accumulators to double outstanding gathers and hide WMMA
// RAW latency. Loss folded per-lane, wave-reduced, atomically accumulated.
__global__ __launch_bounds__(128) void mp2v_loss_kernel(
    const float* __restrict__ emb,
    const int*   __restrict__ walks,
    long nRows, float sgn, float scale,
    float* __restrict__ out)
{
    const int lane = threadIdx.x & 31;
    const int wave = threadIdx.x >> 5;
    const long tile    = (long)blockIdx.x * WAVES_PER_BLOCK + wave;
    const long rowBase = tile * ROWS_PER_WAVE;
    if (rowBase >= nRows) return;   // wave-uniform: EXEC stays all-ones inside

    // This lane serves matrix row/column m; khalf selects the K pair:
    //   lanes 0-15 hold dims {k0, k0+1}, lanes 16-31 hold dims {k0+2, k0+3}
    const int m     = lane & 15;
    const int khalf = lane >> 4;

    long rowA = rowBase + m;
    if (rowA >= nRows) rowA = nRows - 1;   // clamp (counts divide evenly anyway)

    const long wbase = rowA * CTX;

    // Preload ALL walk indices up front: one load clause, no per-context
    // dependent index-fetch latency bubbles later.
    int cidx[CTX];
    #pragma unroll
    for (int c = 0; c < CTX; ++c)
        cidx[c] = walks[wbase + c];

    const float* pa = emb + (size_t)cidx[0] * DIM + (khalf << 1);

    // Prefetch upcoming context embedding rows (global_prefetch_b8); each lane
    // pulls a different half of its row's cachelines.
    #pragma unroll
    for (int c = 3; c < CTX; ++c)
        __builtin_prefetch(emb + (size_t)cidx[c] * DIM + (size_t)(khalf << 6), 0, 0);

    // Cache all 32 K-chunks of the start vector in VGPRs (A is reused 6x).
    v2f Areg[32];
    #pragma unroll
    for (int i = 0; i < 32; ++i)
        Areg[i] = *(const v2f*)(pa + 4 * i);

    // Diagonal ownership: diag(i) lives at {lane=i, acc[i]} for i<8,
    // and {lane=i+16, acc[i-8]} for i>=8.
    const int  sel    = (lane < 8) ? lane : (lane - 24);
    const long ownRow = rowBase + ((lane < 8) ? lane : (lane - 16));
    const bool valid  = ((lane < 8) || (lane >= 24)) && (ownRow < nRows);

    float lanesum = 0.0f;

    // Contexts 1..6 in pairs: two independent accumulator chains per pass.
    #pragma unroll
    for (int cp = 1; cp < CTX; cp += 2) {
        const float* pb0 = emb + (size_t)cidx[cp]     * DIM + (khalf << 1);
        const float* pb1 = emb + (size_t)cidx[cp + 1] * DIM + (khalf << 1);

        v8f acc0 = {};
        v8f acc1 = {};
        #pragma unroll
        for (int k = 0; k < 32; ++k) {
            v2f b0 = *(const v2f*)(pb0 + 4 * k);
            v2f b1 = *(const v2f*)(pb1 + 4 * k);
            acc0 = __builtin_amdgcn_wmma_f32_16x16x4_f32(
                false, Areg[k], false, b0, (short)0, acc0, false, false);
            acc1 = __builtin_amdgcn_wmma_f32_16x16x4_f32(
                false, Areg[k], false, b1, (short)0, acc1, false, false);
        }

        // Branchless diagonal extraction for both contexts.
        float diag0 = 0.0f, diag1 = 0.0f;
        #pragma unroll
        for (int i = 0; i < 8; ++i) {
            diag0 = (sel == i) ? acc0[i] : diag0;
            diag1 = (sel == i) ? acc1[i] : diag1;
        }

        // pos: -log(sigmoid(x)+eps); neg: -log(1-sigmoid(x)+eps) = -log(sigmoid(-x)+eps)
        const float z0   = valid ? (sgn * diag0) : 0.0f;
        const float z1   = valid ? (sgn * diag1) : 0.0f;
        const float sig0 = 1.0f / (1.0f + __expf(-z0));
        const float sig1 = 1.0f / (1.0f + __expf(-z1));
        const float l0   = -__logf(sig0 + 1e-15f);
        const float l1   = -__logf(sig1 + 1e-15f);
        lanesum += valid ? (l0 + l1) : 0.0f;
    }

    // wave32 reduction, then one atomic per wave.
    #pragma unroll
    for (int off = 16; off > 0; off >>= 1)
        lanesum += __shfl_down(lanesum, off, 32);
    if (lane == 0)
        atomicAdd(out, lanesum * scale);
}

extern "C" void kernel_launch(void* const* d_in, const int* in_sizes, int n_in,
                              void* d_out, int out_size, void* d_ws, size_t ws_size,
                              hipStream_t stream) {
    const float* emb = (const float*)d_in[0];
    const int*   pos = (const int*)d_in[1];
    const int*   neg = (const int*)d_in[2];
    float* out = (float*)d_out;

    const long nPos = (long)in_sizes[1] / CTX;   // 200000
    const long nNeg = (long)in_sizes[2] / CTX;   // 1000000

    mp2v_zero_kernel<<<1, 1, 0, stream>>>(out);

    const long wavesPos = (nPos + ROWS_PER_WAVE - 1) / ROWS_PER_WAVE;
    const int  blkPos   = (int)((wavesPos + WAVES_PER_BLOCK - 1) / WAVES_PER_BLOCK);
    mp2v_loss_kernel<<<blkPos, 32 * WAVES_PER_BLOCK, 0, stream>>>(
        emb, pos, nPos, 1.0f, 1.0f / (float)(nPos * (CTX - 1)), out);

    const long wavesNeg = (nNeg + ROWS_PER_WAVE - 1) / ROWS_PER_WAVE;
    const int  blkNeg   = (int)((wavesNeg + WAVES_PER_BLOCK - 1) / WAVES_PER_BLOCK);
    mp2v_loss_kernel<<<blkNeg, 32 * WAVES_PER_BLOCK, 0, stream>>>(
        emb, neg, nNeg, -1.0f, 1.0f / (float)(nNeg * (CTX - 1)), out);
}